// PhraseAttentionExtractor_86895778333343
// MI455X (gfx1250) — compile-verified
//
#include <hip/hip_runtime.h>
#include <hip/hip_bf16.h>
#include <stdint.h>

#define BB 8
#define LL 512
#define HH 768
#define WW 5
#define KK 32
#define PP 256
#define NC 2304          // 3*H combined-weight columns
#define NEGV (-1e30f)

typedef __attribute__((ext_vector_type(2))) float v2f;
typedef __attribute__((ext_vector_type(8))) float v8f;

// ---------------- K2: build Wc = [W1a - W1d | W1b + W1d | W1c] (768 x 2304)
__global__ void build_wc_kernel(const float* __restrict__ w1, float* __restrict__ Wc) {
  int idx = blockIdx.x * 256 + threadIdx.x;          // 768*2304 elements
  int k = idx / NC;
  int n = idx % NC;
  float v;
  if (n < HH)            v = w1[k * HH + n] - w1[(3 * HH + k) * HH + n];
  else if (n < 2 * HH)   v = w1[(HH + k) * HH + (n - HH)] + w1[(3 * HH + k) * HH + (n - HH)];
  else                   v = w1[(2 * HH + k) * HH + (n - 2 * HH)];
  Wc[idx] = v;
}

// ---------------- K3: S = hidden(4096x768) @ Wc(768x2304), f32 WMMA 16x16x4
// Double-buffered via CDNA5 async global->LDS copies (ASYNCcnt).
__global__ __launch_bounds__(256) void gemm_wmma_kernel(const float* __restrict__ X,
                                                        const float* __restrict__ Wc,
                                                        float* __restrict__ S) {
  // rows padded so every 16B async chunk is 16B-aligned in LDS and the
  // column-ish A reads spread across banks (80B row stride -> 16 banks).
  __shared__ float Xs[2][64][20];
  __shared__ float Ws[2][16][132];
  int t    = threadIdx.x;
  int lane = t & 31;
  int wv   = t >> 5;               // 8 waves
  int gm0  = blockIdx.y * 64;
  int gn0  = blockIdx.x * 128;
  int m0w  = (wv & 3) * 16;        // wave's row tile
  int n0w  = (wv >> 2) * 64;       // wave's 64-wide col strip
  int ln   = lane & 15;
  int kd   = (lane >> 4) * 2;      // lanes 16-31 hold K+2,K+3 halves

  v8f acc[4];
#pragma unroll
  for (int q = 0; q < 4; ++q) acc[q] = (v8f){0, 0, 0, 0, 0, 0, 0, 0};

  // per-thread async-copy assignment (16B chunks)
  int xr = t >> 2;                 // 0..63
  int xc = (t & 3) * 4;            // 0,4,8,12
  int wk = t >> 5;                 // 0..7
  int wn = (t & 31) * 4;           // 0..124

  const float* gx  = X  + (size_t)(gm0 + xr) * HH + xc;        // + k0
  const float* gw0 = Wc + (size_t)wk * NC + gn0 + wn;          // + k0*NC
  const float* gw1 = Wc + (size_t)(wk + 8) * NC + gn0 + wn;    // + k0*NC

  auto issue_tile = [&](int buf, int k0) {
    uint32_t lx  = (uint32_t)(uintptr_t)&Xs[buf][xr][xc];
    uint32_t lw0 = (uint32_t)(uintptr_t)&Ws[buf][wk][wn];
    uint32_t lw1 = (uint32_t)(uintptr_t)&Ws[buf][wk + 8][wn];
    const float* px  = gx + k0;
    const float* pw0 = gw0 + (size_t)k0 * NC;
    const float* pw1 = gw1 + (size_t)k0 * NC;
    asm volatile("global_load_async_to_lds_b128 %0, %1, off"
                 :: "v"(lx),  "v"(px)  : "memory");
    asm volatile("global_load_async_to_lds_b128 %0, %1, off"
                 :: "v"(lw0), "v"(pw0) : "memory");
    asm volatile("global_load_async_to_lds_b128 %0, %1, off"
                 :: "v"(lw1), "v"(pw1) : "memory");
  };

  issue_tile(0, 0);
  asm volatile("s_wait_asynccnt 0x0" ::: "memory");
  __syncthreads();

  int buf = 0;
  for (int k0 = 0; k0 < HH; k0 += 16) {
    if (k0 + 16 < HH) issue_tile(buf ^ 1, k0 + 16);   // prefetch next tile
#pragma unroll
    for (int kk = 0; kk < 16; kk += 4) {
      v2f a;
      a.x = Xs[buf][m0w + ln][kk + kd];
      a.y = Xs[buf][m0w + ln][kk + kd + 1];
#pragma unroll
      for (int tt = 0; tt < 4; ++tt) {
        v2f bb;
        bb.x = Ws[buf][kk + kd][n0w + tt * 16 + ln];
        bb.y = Ws[buf][kk + kd + 1][n0w + tt * 16 + ln];
        acc[tt] = __builtin_amdgcn_wmma_f32_16x16x4_f32(
            false, a, false, bb, (short)0, acc[tt], false, false);
      }
    }
    asm volatile("s_wait_asynccnt 0x0" ::: "memory");  // next tile landed
    __syncthreads();                                    // all waves' copies visible
    buf ^= 1;
  }

  int rbase = gm0 + m0w + (lane >> 4) * 8;
#pragma unroll
  for (int tt = 0; tt < 4; ++tt) {
    int col = gn0 + n0w + tt * 16 + ln;
#pragma unroll
    for (int r = 0; r < 8; ++r)
      S[(size_t)(rbase + r) * NC + col] = acc[tt][r];
  }
}

// ---------------- K1: PH = [0; cumsum_L(hidden)]  (B,(L+1),H)
__global__ void cumsum_hidden_kernel(const float* __restrict__ hid, float* __restrict__ PH) {
  int col = blockIdx.x * 256 + threadIdx.x;   // B*H columns
  int b = col / HH, hh = col % HH;
  const float* src = hid + (size_t)b * LL * HH + hh;
  float* dst = PH + (size_t)b * (LL + 1) * HH + hh;
  float s = 0.f;
  dst[0] = 0.f;
  for (int i = 0; i < LL; ++i) { s += src[(size_t)i * HH]; dst[(size_t)(i + 1) * HH] = s; }
}

// ---------------- K4: PSC = [0; cumsum_L(S[:, 2H:3H])]  (B,(L+1),H)
__global__ void cumsum_sc_kernel(const float* __restrict__ S, float* __restrict__ PSC) {
  int col = blockIdx.x * 256 + threadIdx.x;
  int b = col / HH, n = col % HH;
  const float* src = S + (size_t)b * LL * NC + 2 * HH + n;
  float* dst = PSC + (size_t)b * (LL + 1) * HH + n;
  float s = 0.f;
  dst[0] = 0.f;
  for (int i = 0; i < LL; ++i) { s += src[(size_t)i * NC]; dst[(size_t)(i + 1) * HH] = s; }
}

// ---------------- K5: per-span h -> LN -> relu -> dot(w2) -> masked score
__global__ __launch_bounds__(256) void score_kernel(
    const float* __restrict__ S, const float* __restrict__ PSC,
    const float* __restrict__ b1, const float* __restrict__ g1,
    const float* __restrict__ beta1, const float* __restrict__ w2,
    const float* __restrict__ b2, const int* __restrict__ attn,
    float* __restrict__ scores) {
  __shared__ float red1[256];
  __shared__ float red2[256];
  int b = blockIdx.y, i = blockIdx.x, t = threadIdx.x;
  const float* SArow = S + (size_t)(b * LL + i) * NC;
  const float* PCi   = PSC + ((size_t)b * (LL + 1) + i) * HH;
  float sa[3], pci[3];
#pragma unroll
  for (int e = 0; e < 3; ++e) { int idx = t + e * 256; sa[e] = SArow[idx]; pci[e] = PCi[idx]; }

  for (int w = 0; w < WW; ++w) {
    int j = min(i + w, LL - 1);
    const float* SBrow = S + (size_t)(b * LL + j) * NC + HH;
    const float* PCj   = PSC + ((size_t)b * (LL + 1) + j + 1) * HH;
    float inv = 1.0f / (float)(w + 1);
    float h[3]; float s = 0.f, s2 = 0.f;
#pragma unroll
    for (int e = 0; e < 3; ++e) {
      int idx = t + e * 256;
      float v = sa[e] + SBrow[idx] + (PCj[idx] - pci[e]) * inv + b1[idx];
      h[e] = v; s += v; s2 += v * v;
    }
    red1[t] = s; red2[t] = s2; __syncthreads();
    for (int sft = 128; sft > 0; sft >>= 1) {
      if (t < sft) { red1[t] += red1[t + sft]; red2[t] += red2[t + sft]; }
      __syncthreads();
    }
    float mean = red1[0] * (1.0f / HH);
    float var  = red2[0] * (1.0f / HH) - mean * mean;
    float rstd = rsqrtf(var + 1e-5f);
    float dot = 0.f;
#pragma unroll
    for (int e = 0; e < 3; ++e) {
      int idx = t + e * 256;
      float v = (h[e] - mean) * rstd * g1[idx] + beta1[idx];
      v = fmaxf(v, 0.f);
      dot += v * w2[idx];
    }
    __syncthreads();
    red1[t] = dot; __syncthreads();
    for (int sft = 128; sft > 0; sft >>= 1) {
      if (t < sft) red1[t] += red1[t + sft];
      __syncthreads();
    }
    if (t == 0) {
      float sc = red1[0] + b2[0];
      bool ok = (i + w) < LL;
      if (ok) for (int kk = 0; kk <= w; ++kk) ok = ok && (attn[b * LL + i + kk] != 0);
      scores[(size_t)b * LL * WW + i * WW + w] = ok ? sc : NEGV;
    }
    __syncthreads();
  }
}

// ---------------- K6: per-batch top-32 (iterative argmax), writes masks/scores/spans
__global__ __launch_bounds__(256) void topk_kernel(
    const float* __restrict__ scores, float* __restrict__ topS, int* __restrict__ topI,
    float* __restrict__ o_mask, float* __restrict__ o_scr, float* __restrict__ o_spans) {
  __shared__ float sc[LL * WW];
  __shared__ float rv[256];
  __shared__ int   ri[256];
  int b = blockIdx.x, t = threadIdx.x;
  for (int u = t; u < LL * WW; u += 256) sc[u] = scores[(size_t)b * LL * WW + u];
  __syncthreads();
  for (int sel = 0; sel < KK; ++sel) {
    float bv = -3.0e38f; int bi = 1 << 30;
    for (int u = t; u < LL * WW; u += 256) {
      float v = sc[u];
      if (v > bv || (v == bv && u < bi)) { bv = v; bi = u; }
    }
    rv[t] = bv; ri[t] = bi; __syncthreads();
    for (int sft = 128; sft > 0; sft >>= 1) {
      if (t < sft) {
        if (rv[t + sft] > rv[t] || (rv[t + sft] == rv[t] && ri[t + sft] < ri[t])) {
          rv[t] = rv[t + sft]; ri[t] = ri[t + sft];
        }
      }
      __syncthreads();
    }
    if (t == 0) {
      topS[b * KK + sel] = rv[0];
      topI[b * KK + sel] = ri[0];
      sc[ri[0]] = -3.0e38f;
    }
    __syncthreads();
  }
  if (t < KK) {
    float ts = topS[b * KK + t];
    int   ti = topI[b * KK + t];
    bool m = ts > (NEGV * 0.5f);
    o_mask[b * KK + t] = m ? 1.0f : 0.0f;
    o_scr[b * KK + t]  = m ? ts : -10.0f;
    int si = ti / WW, sw = ti % WW;
    o_spans[(b * KK + t) * 2 + 0] = m ? (float)si : 0.0f;
    o_spans[(b * KK + t) * 2 + 1] = m ? (float)(si + sw) : 0.0f;
  }
}

// ---------------- K7: gather span-mean embeddings for top-k
__global__ void embs_kernel(const float* __restrict__ PH, const float* __restrict__ topS,
                            const int* __restrict__ topI, float* __restrict__ embs) {
  int bk = blockIdx.x;            // 0..255
  int b = bk >> 5;
  int t = threadIdx.x;
  int ti = topI[bk];
  bool m = topS[bk] > (NEGV * 0.5f);
  int i = ti / WW, w = ti % WW;
  int j = min(i + w, LL - 1);
  const float* Pj = PH + ((size_t)b * (LL + 1) + j + 1) * HH;
  const float* Pi = PH + ((size_t)b * (LL + 1) + i) * HH;
  float inv = 1.0f / (float)(w + 1);
#pragma unroll
  for (int e = 0; e < 3; ++e) {
    int idx = t + e * 256;
    embs[(size_t)bk * HH + idx] = m ? (Pj[idx] - Pi[idx]) * inv : 0.0f;
  }
}

// ---------------- K8: gate logits = tanh(embs@wg1+bg1)@wg2 + bg2  (masked)
__global__ __launch_bounds__(256) void gate_kernel(
    const float* __restrict__ embs, const float* __restrict__ wg1,
    const float* __restrict__ bg1, const float* __restrict__ wg2,
    const float* __restrict__ bg2, const float* __restrict__ topS,
    float* __restrict__ gate) {
  __shared__ float er[HH];
  __shared__ float red[256];
  int bk = blockIdx.x, t = threadIdx.x;
#pragma unroll
  for (int e = 0; e < 3; ++e) er[t + e * 256] = embs[(size_t)bk * HH + t + e * 256];
  __syncthreads();
  float acc = 0.f;
  for (int u = t; u < HH / 2; u += 256) {
    float d = bg1[u];
    for (int hh = 0; hh < HH; ++hh) d += er[hh] * wg1[(size_t)hh * (HH / 2) + u];
    acc += tanhf(d) * wg2[u];
  }
  red[t] = acc; __syncthreads();
  for (int sft = 128; sft > 0; sft >>= 1) {
    if (t < sft) red[t] += red[t + sft];
    __syncthreads();
  }
  if (t == 0) {
    bool m = topS[bk] > (NEGV * 0.5f);
    gate[bk] = m ? (red[0] + bg2[0]) : -1e9f;
  }
}

// ---------------- K9: softmax over K per batch -> phrase_attention
__global__ void softmax_kernel(const float* __restrict__ gate, float* __restrict__ o_attn) {
  __shared__ float g[KK];
  __shared__ float ex[KK];
  int b = blockIdx.x, t = threadIdx.x;
  g[t] = gate[b * KK + t];
  __syncthreads();
  float mx = -3.0e38f;
  for (int u = 0; u < KK; ++u) mx = fmaxf(mx, g[u]);
  ex[t] = __expf(g[t] - mx);
  __syncthreads();
  float sm = 0.f;
  for (int u = 0; u < KK; ++u) sm += ex[u];
  o_attn[b * KK + t] = ex[t] / sm;
}

// ---------------- K10: phrase_embeds = LN(embs@wp + bp; gout, bout)
__global__ __launch_bounds__(256) void proj_kernel(
    const float* __restrict__ embs, const float* __restrict__ wp,
    const float* __restrict__ bp, const float* __restrict__ gout,
    const float* __restrict__ bout, float* __restrict__ o_embeds) {
  __shared__ float er[HH];
  __shared__ float red1[256];
  __shared__ float red2[256];
  int bk = blockIdx.x, t = threadIdx.x;
#pragma unroll
  for (int e = 0; e < 3; ++e) er[t + e * 256] = embs[(size_t)bk * HH + t + e * 256];
  __syncthreads();
  float y = bp[t];
  for (int hh = 0; hh < HH; ++hh) y += er[hh] * wp[(size_t)hh * PP + t];
  red1[t] = y; red2[t] = y * y; __syncthreads();
  for (int sft = 128; sft > 0; sft >>= 1) {
    if (t < sft) { red1[t] += red1[t + sft]; red2[t] += red2[t + sft]; }
    __syncthreads();
  }
  float mean = red1[0] * (1.0f / PP);
  float var  = red2[0] * (1.0f / PP) - mean * mean;
  float rstd = rsqrtf(var + 1e-5f);
  o_embeds[(size_t)bk * PP + t] = (y - mean) * rstd * gout[t] + bout[t];
}

extern "C" void kernel_launch(void* const* d_in, const int* in_sizes, int n_in,
                              void* d_out, int out_size, void* d_ws, size_t ws_size,
                              hipStream_t stream) {
  (void)in_sizes; (void)n_in; (void)out_size; (void)ws_size;
  const float* hidden = (const float*)d_in[0];
  const int*   attn   = (const int*)d_in[1];
  const float* w1     = (const float*)d_in[2];
  const float* b1     = (const float*)d_in[3];
  const float* g1     = (const float*)d_in[4];
  const float* beta1  = (const float*)d_in[5];
  const float* w2     = (const float*)d_in[6];
  const float* b2     = (const float*)d_in[7];
  const float* wp     = (const float*)d_in[8];
  const float* bp     = (const float*)d_in[9];
  const float* wg1    = (const float*)d_in[10];
  const float* bg1    = (const float*)d_in[11];
  const float* wg2    = (const float*)d_in[12];
  const float* bg2    = (const float*)d_in[13];
  const float* gout   = (const float*)d_in[14];
  const float* bout   = (const float*)d_in[15];

  // workspace layout (floats)
  float* Wc     = (float*)d_ws;                              // 768*2304
  float* Smat   = Wc + (size_t)HH * NC;                      // 4096*2304
  float* PH     = Smat + (size_t)BB * LL * NC;               // 8*513*768
  float* PSC    = PH + (size_t)BB * (LL + 1) * HH;           // 8*513*768
  float* scores = PSC + (size_t)BB * (LL + 1) * HH;          // 8*2560
  float* topS   = scores + (size_t)BB * LL * WW;             // 256
  int*   topI   = (int*)(topS + BB * KK);                    // 256
  float* embs   = (float*)(topI + BB * KK);                  // 256*768
  float* gate   = embs + (size_t)BB * KK * HH;               // 256

  // output layout (floats, concatenated in return order)
  float* out       = (float*)d_out;
  float* o_embeds  = out;                                    // 8*32*256
  float* o_mask    = o_embeds + (size_t)BB * KK * PP;        // 256
  float* o_attn    = o_mask + BB * KK;                       // 256
  float* o_scr     = o_attn + BB * KK;                       // 256
  float* o_spans   = o_scr + BB * KK;                        // 512

  build_wc_kernel<<<(HH * NC) / 256, 256, 0, stream>>>(w1, Wc);
  cumsum_hidden_kernel<<<(BB * HH) / 256, 256, 0, stream>>>(hidden, PH);
  gemm_wmma_kernel<<<dim3(NC / 128, (BB * LL) / 64), 256, 0, stream>>>(hidden, Wc, Smat);
  cumsum_sc_kernel<<<(BB * HH) / 256, 256, 0, stream>>>(Smat, PSC);
  score_kernel<<<dim3(LL, BB), 256, 0, stream>>>(Smat, PSC, b1, g1, beta1, w2, b2, attn, scores);
  topk_kernel<<<BB, 256, 0, stream>>>(scores, topS, topI, o_mask, o_scr, o_spans);
  embs_kernel<<<BB * KK, 256, 0, stream>>>(PH, topS, topI, embs);
  gate_kernel<<<BB * KK, 256, 0, stream>>>(embs, wg1, bg1, wg2, bg2, topS, gate);
  softmax_kernel<<<BB, KK, 0, stream>>>(gate, o_attn);
  proj_kernel<<<BB * KK, 256, 0, stream>>>(embs, wp, bp, gout, bout, o_embeds);
}